// PriorConv2d_25709674234115
// MI455X (gfx1250) — compile-verified
//
#include <hip/hip_runtime.h>
#include <hip/hip_bf16.h>
#include <math.h>

#define HH 384
#define WW 384
#define NB 8
#define HWSZ (HH*WW)          /* 147456 */
#define NPIX (NB*HWSZ)        /* 1179648 */
#define KTAPS 37
#define RAD 18

typedef __attribute__((ext_vector_type(2))) float v2f;
typedef __attribute__((ext_vector_type(8))) float v8f;
typedef __attribute__((ext_vector_type(4))) unsigned int u32x4;
typedef __attribute__((ext_vector_type(8))) int i32x8;
typedef __attribute__((ext_vector_type(4))) int i32x4;

__device__ __forceinline__ v8f wmma_k4(v2f a, v2f b, v8f c) {
  // D(16x16,f32) = A(16x4,f32) x B(4x16,f32) + C  -- exact fp32 matrix op
  return __builtin_amdgcn_wmma_f32_16x16x4_f32(false, a, false, b, (short)0, c, false, false);
}

// ---------------- workspace layout (floats) ----------------
#define XG_OFF   0
#define XG_SZ    (NB*3*HWSZ)
#define BUFA_OFF (XG_OFF + XG_SZ)
#define BUFA_SZ  (NB*16*HWSZ)
#define BUFB_OFF (BUFA_OFF + BUFA_SZ)
#define BUFB_SZ  (NB*16*HWSZ)
#define EB_OFF   (BUFB_OFF + BUFB_SZ)
#define EB_SZ    (NB*3*HWSZ)
#define PART_OFF (EB_OFF + EB_SZ)
#define PART_SZ  (NB*576)
#define TAPS_OFF (PART_OFF + PART_SZ)
#define TAPS_SZ  (NB*KTAPS)
#define MM_OFF   (TAPS_OFF + TAPS_SZ)
#define MM_SZ    (64*8)
#define BIAS_OFF (MM_OFF + MM_SZ)

// ---------------- xg = gcm @ x (3x3 color matrix) ----------------
__global__ void k_xg(const float* __restrict__ x, const float* __restrict__ gcm,
                     float* __restrict__ xg) {
  int p = blockIdx.x * blockDim.x + threadIdx.x;
  if (p >= NPIX) return;
  int b = p / HWSZ, rem = p - b * HWSZ;
  float x0 = x[(size_t)(b*3+0)*HWSZ + rem];
  float x1 = x[(size_t)(b*3+1)*HWSZ + rem];
  float x2 = x[(size_t)(b*3+2)*HWSZ + rem];
  #pragma unroll
  for (int i = 0; i < 3; ++i)
    xg[(size_t)(b*3+i)*HWSZ + rem] = gcm[i*3+0]*x0 + gcm[i*3+1]*x1 + gcm[i*3+2]*x2;
}

// ---------------- implicit-GEMM 3x3 conv -> 16 channels via f32 WMMA ----------------
// Block = 16x16 pixel tile. out(16ch x 16pix row) = W(16 x K) x patch(K x 16), K = CIN*9,
// padded to a multiple of 4 and evaluated as chained V_WMMA_F32_16X16X4_F32.
// ACT: 1 = silu, 2 = relu.
template<int CIN, int ACT>
__global__ void __launch_bounds__(256)
k_convw(const float* __restrict__ in, const float* __restrict__ w,
        const float* __restrict__ bias, float* __restrict__ out) {
  constexpr int KTOT = CIN * 9;
  constexpr int KPAD = (KTOT + 3) & ~3;
  __shared__ float wl[16 * KPAD];            // A matrix: 16 out-ch x KPAD
  __shared__ float bl[16];
  __shared__ float tin[CIN][18][18];         // input tile + halo
  __shared__ unsigned int pk[KPAD / 2];      // packed patch LDS offsets (2 x u16)
  int tid = threadIdx.x;

  if constexpr (KTOT == KPAD) {
    // Weights are one contiguous 16*K*4-byte tile: stage with the Tensor Data Mover.
    if ((tid >> 5) == 0) {
      unsigned int lds = (unsigned int)(uintptr_t)&wl[0];
      unsigned long long ga = (unsigned long long)(uintptr_t)w;
      u32x4 g0;
      g0[0] = 1u;                                          // count=1 (valid user D#)
      g0[1] = lds;                                         // lds_addr
      g0[2] = (unsigned int)(ga & 0xFFFFFFFFull);          // global_addr[31:0]
      g0[3] = (unsigned int)((ga >> 32) & 0x01FFFFFFull)   // global_addr[56:32]
            | (2u << 30);                                  // type = 2 ("image")
      i32x8 g1;
      g1[0] = (2 << 16);                                   // data_size = 4 bytes
      g1[1] = (int)(((16 * KTOT) & 0xFFFF) << 16);         // tensor_dim0[15:0]
      g1[2] = (int)(((16 * KTOT) >> 16) | (1 << 16));      // tensor_dim0[31:16] | tensor_dim1=1
      g1[3] = (int)(((16 * KTOT) & 0xFFFF) << 16);         // tile_dim0 = 16*K
      g1[4] = 1;                                           // tile_dim1 = 1
      g1[5] = 16 * KTOT;                                   // tensor_dim0_stride[31:0]
      g1[6] = 0;
      g1[7] = 0;
      i32x4 z4 = {0, 0, 0, 0};
      i32x8 z8 = {0, 0, 0, 0, 0, 0, 0, 0};
      __builtin_amdgcn_tensor_load_to_lds(g0, g1, z4, z4, z8, 0);  // 6-arg (clang-23) form
      __builtin_amdgcn_s_wait_tensorcnt((short)0);
    }
  } else {
    // K padded (27 -> 28): repack with zero fill.
    for (int idx = tid; idx < 16 * KPAD; idx += 256) {
      int o = idx / KPAD, k = idx - o * KPAD;
      wl[idx] = (k < KTOT) ? w[o * KTOT + k] : 0.f;
    }
  }
  if (tid < 16) bl[tid] = bias[tid];
  if (tid < KPAD / 2) {
    int k0 = 2 * tid, k1 = 2 * tid + 1;
    unsigned int o0 = (k0 < KTOT) ? (unsigned)((k0/9)*324 + ((k0%9)/3)*18 + (k0%3)) : 0u;
    unsigned int o1 = (k1 < KTOT) ? (unsigned)((k1/9)*324 + ((k1%9)/3)*18 + (k1%3)) : 0u;
    pk[tid] = o0 | (o1 << 16);
  }

  // Stage input tile (with zero-padded halo) into LDS.
  int b  = blockIdx.x / 576;
  int tr = blockIdx.x % 576;
  int ty = tr / 24, tx = tr - ty * 24;
  int h0 = ty * 16, w0 = tx * 16;
  for (int idx = tid; idx < CIN * 324; idx += 256) {
    int c = idx / 324, r = idx - c * 324;
    int yy = r / 18, xx = r - yy * 18;
    int hh = h0 + yy - 1, ww2 = w0 + xx - 1;
    float v = 0.f;
    if (hh >= 0 && hh < HH && ww2 >= 0 && ww2 < WW)
      v = in[(size_t)(b * CIN + c) * HWSZ + hh * WW + ww2];
    tin[c][yy][xx] = v;
  }
  __syncthreads();

  // WMMA: wave handles 2 pixel rows; one 16x16 (ch x pix) D tile per row.
  int wv = tid >> 5, l = tid & 31, hlf = l >> 4, pl = l & 15;
  const float* wrow = &wl[pl * KPAD + hlf * 2];
  #pragma unroll
  for (int g = 0; g < 2; ++g) {
    int y = wv * 2 + g;
    const float* pbase = &tin[0][0][0] + y * 18 + pl;
    v8f cc;
    #pragma unroll
    for (int rr = 0; rr < 8; ++rr) cc[rr] = bl[hlf * 8 + rr];
    #pragma unroll
    for (int k0 = 0; k0 < KPAD; k0 += 4) {
      unsigned int pr = pk[(k0 >> 1) + hlf];
      v2f af, bf;
      af.x = wrow[k0];  af.y = wrow[k0 + 1];
      bf.x = pbase[pr & 0xFFFFu];
      bf.y = pbase[pr >> 16];
      cc = wmma_k4(af, bf, cc);
    }
    #pragma unroll
    for (int rr = 0; rr < 8; ++rr) {
      int m = hlf * 8 + rr;
      float v = cc[rr];
      if (ACT == 1) v = v / (1.f + __expf(-v));   // silu
      else if (ACT == 2) v = fmaxf(v, 0.f);       // relu
      out[(size_t)(b * 16 + m) * HWSZ + (h0 + y) * WW + (w0 + pl)] = v;
    }
  }
}

// ---------------- c2: 3x3 conv 16->1 fused with per-batch block reduction ----------------
__global__ void k_c2_reduce(const float* __restrict__ in, const float* __restrict__ w,
                            const float* __restrict__ bias, float* __restrict__ part) {
  __shared__ float wl[144];
  __shared__ float red[256];
  if (threadIdx.x < 144) wl[threadIdx.x] = w[threadIdx.x];
  __syncthreads();
  int b = blockIdx.x / 576;
  int rem = (blockIdx.x % 576) * 256 + threadIdx.x;
  int h = rem / WW, wx = rem - h * WW;
  float acc = bias[0];
  for (int c = 0; c < 16; ++c) {
    const float* inp = in + (size_t)(b*16 + c) * HWSZ;
    #pragma unroll
    for (int ky = 0; ky < 3; ++ky) {
      int hh = h + ky - 1; if (hh < 0 || hh >= HH) continue;
      #pragma unroll
      for (int kx = 0; kx < 3; ++kx) {
        int ww2 = wx + kx - 1; if (ww2 < 0 || ww2 >= WW) continue;
        acc += wl[c*9 + ky*3 + kx] * inp[hh*WW + ww2];
      }
    }
  }
  red[threadIdx.x] = acc;
  __syncthreads();
  for (int s = 128; s > 0; s >>= 1) {
    if (threadIdx.x < s) red[threadIdx.x] += red[threadIdx.x + s];
    __syncthreads();
  }
  if (threadIdx.x == 0) part[blockIdx.x] = red[0];
}

// ---------------- per-batch scale -> normalized separable Gaussian taps ----------------
__global__ void k_taps(const float* __restrict__ part, float* __restrict__ taps) {
  __shared__ float g[NB][KTAPS];
  __shared__ float stdv[NB];
  __shared__ float fr[NB];
  __shared__ float ssum[NB];
  int t = threadIdx.x;
  if (t < NB) {
    float s = 0.f;
    for (int i = 0; i < 576; ++i) s += part[t*576 + i];
    float mean = s / (float)HWSZ;
    float sc = fminf(2.5f, fmaxf(-2.5f, mean));    // clip
    float sd = exp2f(sc);                          // std = 2^scale
    stdv[t] = sd;
    fr[t] = ceilf(3.f*sd + 0.5f);
  }
  __syncthreads();
  for (int idx = t; idx < NB*KTAPS; idx += blockDim.x) {
    int b = idx / KTAPS, i = idx - b*KTAPS;
    float xx = (float)(i - RAD);
    float q = xx / stdv[b];
    g[b][i] = (fabsf(xx) <= fr[b]) ? __expf(-0.5f*q*q) : 0.f;
  }
  __syncthreads();
  if (t < NB) {
    float s = 0.f;
    for (int i = 0; i < KTAPS; ++i) s += g[t][i];
    ssum[t] = s;
  }
  __syncthreads();
  for (int idx = t; idx < NB*KTAPS; idx += blockDim.x) {
    int b = idx / KTAPS, i = idx - b*KTAPS;
    taps[b*KTAPS + i] = g[b][i] / ssum[b];
  }
}

// ---------------- horizontal blur: one row per block (full row fits LDS) ----------------
__global__ void k_blur_h(const float* __restrict__ src, const float* __restrict__ taps,
                         float* __restrict__ dst) {
  __shared__ float row[WW + 2*RAD];
  __shared__ float tp[KTAPS];
  int r = blockIdx.x;                 // over NB*3*HH rows
  int h = r % HH;
  int c = (r / HH) % 3;
  int b = r / (3*HH);
  int t = threadIdx.x;                // 0..383
  row[RAD + t] = src[(size_t)(b*3 + c)*HWSZ + h*WW + t];
  if (t < RAD) { row[t] = 0.f; row[WW + RAD + t] = 0.f; }   // zero pad (FMAX halo)
  if (t < KTAPS) tp[t] = taps[b*KTAPS + t];
  __syncthreads();
  float acc = 0.f;
  #pragma unroll
  for (int i = 0; i < KTAPS; ++i) acc += tp[i] * row[t + i];
  dst[(size_t)(b*3 + c)*HWSZ + h*WW + t] = acc;
}

// ---------------- vertical blur: coalesced column sweep ----------------
__global__ void k_blur_v(const float* __restrict__ src, const float* __restrict__ taps,
                         float* __restrict__ dst) {
  __shared__ float tp[KTAPS];
  int p = blockIdx.x * blockDim.x + threadIdx.x;   // over NB*3*HWSZ
  int bc = p / HWSZ;                               // (b*3+c): constant per block
  int rem = p - bc * HWSZ;
  int b = bc / 3;
  if (threadIdx.x < KTAPS) tp[threadIdx.x] = taps[b*KTAPS + threadIdx.x];
  __syncthreads();
  int h = rem / WW, wx = rem - h * WW;
  const float* sp = src + (size_t)bc * HWSZ;
  float acc = 0.f;
  for (int i = 0; i < KTAPS; ++i) {
    int hh = h + i - RAD;
    if (hh >= 0 && hh < HH) acc += tp[i] * sp[hh*WW + wx];
  }
  dst[(size_t)bc*HWSZ + rem] = acc;
}

// ---------------- collapse the 64x64 tail conv into a 64x8 matrix ----------------
__global__ void k_build_M(const float* __restrict__ adj, const float* __restrict__ adj_b,
                          const float* __restrict__ cHw, const float* __restrict__ cHb,
                          const float* __restrict__ cSw, const float* __restrict__ cSb,
                          const float* __restrict__ cRw, const float* __restrict__ cRb,
                          float* __restrict__ M, float* __restrict__ biasv) {
  int o = threadIdx.x;
  if (o >= 64) return;
  float m0=0,m1=0,m2=0,m3=0,m4=0,m5=0,m6=0,m7=0;
  for (int k = 0; k < 16; ++k) {
    float aH = adj[o*64 + k];
    float aS = adj[o*64 + 16 + k];
    float aR = adj[o*64 + 32 + k];          // adj cols 48..63 multiply zeros -> dropped
    m0 += aH * cHw[k];   m1 += aH * cHb[k];
    m2 += aS * cSw[k];   m3 += aS * cSb[k];
    m4 += aR * cRw[k*3+0]; m5 += aR * cRw[k*3+1]; m6 += aR * cRw[k*3+2];
    m7 += aR * cRb[k];
  }
  M[o*8+0]=m0; M[o*8+1]=m1; M[o*8+2]=m2; M[o*8+3]=m3;
  M[o*8+4]=m4; M[o*8+5]=m5; M[o*8+6]=m6; M[o*8+7]=m7;
  biasv[o] = adj_b[o];
}

// ---------------- fused tail: wp2+softmax, Hc/Sc, RGB order, 64x8 WMMA GEMM ----------------
__global__ void __launch_bounds__(256)
k_final(const float* __restrict__ x, const float* __restrict__ wbuf,
        const float* __restrict__ eb, const float* __restrict__ Mg,
        const float* __restrict__ biasg, const float* __restrict__ wp2w,
        const float* __restrict__ wp2b, float* __restrict__ out) {
  __shared__ float tld[256][8];   // per-pixel t-vectors (B matrix source)
  __shared__ float Mld[64][8];    // collapsed 64x8 matrix (A matrix source)
  __shared__ float adjb[64];
  __shared__ float w2l[64];
  __shared__ float b2l[4];
  int tid = threadIdx.x;
  for (int i = tid; i < 512; i += 256) (&Mld[0][0])[i] = Mg[i];
  if (tid < 64) { adjb[tid] = biasg[tid]; w2l[tid] = wp2w[tid]; }
  if (tid < 4)  b2l[tid] = wp2b[tid];
  __syncthreads();

  int p = blockIdx.x * 256 + tid;
  int b = p / HWSZ, rem = p - b * HWSZ;

  // wp2 1x1 conv (16->4) + softmax
  float lg0 = b2l[0], lg1 = b2l[1], lg2 = b2l[2], lg3 = b2l[3];
  #pragma unroll
  for (int k = 0; k < 16; ++k) {
    float v = wbuf[(size_t)(b*16 + k)*HWSZ + rem];
    lg0 += w2l[0*16+k]*v; lg1 += w2l[1*16+k]*v;
    lg2 += w2l[2*16+k]*v; lg3 += w2l[3*16+k]*v;
  }
  float mxl4 = fmaxf(fmaxf(lg0, lg1), fmaxf(lg2, lg3));
  float e0 = __expf(lg0-mxl4), e1 = __expf(lg1-mxl4);
  float e2 = __expf(lg2-mxl4), e3 = __expf(lg3-mxl4);
  float inv = 1.f / (e0+e1+e2+e3);
  float w0 = e0*inv, w1 = e1*inv, w2 = e2*inv;

  // Hc, Sc from blurred E/El/Ell
  float E   = eb[(size_t)(b*3+0)*HWSZ + rem];
  float El  = eb[(size_t)(b*3+1)*HWSZ + rem];
  float Ell = eb[(size_t)(b*3+2)*HWSZ + rem];
  float Hc = atanf(El / (Ell + 1e-4f));
  float Sc = logf((El*El + Ell*Ell) / (E*E + 1e-4f) + 1e-4f);

  // RGB order (first/last argmax/argmin for tie handling)
  float xr = x[(size_t)(b*3+0)*HWSZ + rem];
  float xg = x[(size_t)(b*3+1)*HWSZ + rem];
  float xb = x[(size_t)(b*3+2)*HWSZ + rem];
  int mxf = 0; float mv = xr; if (xg > mv){mv=xg;mxf=1;} if (xb > mv){mv=xb;mxf=2;}
  int mxr = 2; mv = xb;       if (xg > mv){mv=xg;mxr=1;} if (xr > mv){mv=xr;mxr=0;}
  int mnf = 0; mv = xr;       if (xg < mv){mv=xg;mnf=1;} if (xb < mv){mv=xb;mnf=2;}
  int mnr = 2; mv = xb;       if (xg < mv){mv=xg;mnr=1;} if (xr < mv){mv=xr;mnr=0;}
  float R0 = 0.5f*((mxf==0)+(mxr==0)) - 0.5f*((mnf==0)+(mnr==0));
  float R1 = 0.5f*((mxf==1)+(mxr==1)) - 0.5f*((mnf==1)+(mnr==1));
  float R2 = 0.5f*((mxf==2)+(mxr==2)) - 0.5f*((mnf==2)+(mnr==2));

  tld[tid][0] = Hc*w0; tld[tid][1] = w0;
  tld[tid][2] = Sc*w1; tld[tid][3] = w1;
  tld[tid][4] = R0*w2; tld[tid][5] = R1*w2;
  tld[tid][6] = R2*w2; tld[tid][7] = w2;
  __syncthreads();

  // WMMA phase: wave handles 32 pixels = two 16-pixel N tiles; 4 M tiles of out channels.
  int wv = tid >> 5, l = tid & 31, hlf = l >> 4, pl = l & 15;
  #pragma unroll
  for (int grp = 0; grp < 2; ++grp) {
    int trow = wv*32 + grp*16 + pl;
    v2f bf0, bf1;
    bf0.x = tld[trow][hlf*2 + 0];     bf0.y = tld[trow][hlf*2 + 1];
    bf1.x = tld[trow][4 + hlf*2 + 0]; bf1.y = tld[trow][4 + hlf*2 + 1];
    int pp = blockIdx.x*256 + wv*32 + grp*16 + pl;
    int pb = pp / HWSZ, prem = pp - pb*HWSZ;
    #pragma unroll
    for (int mt = 0; mt < 4; ++mt) {
      int arow = mt*16 + pl;
      v2f af0, af1;
      af0.x = Mld[arow][hlf*2 + 0];     af0.y = Mld[arow][hlf*2 + 1];
      af1.x = Mld[arow][4 + hlf*2 + 0]; af1.y = Mld[arow][4 + hlf*2 + 1];
      v8f cc = {0.f,0.f,0.f,0.f,0.f,0.f,0.f,0.f};
      cc = wmma_k4(af0, bf0, cc);   // K = 0..3
      cc = wmma_k4(af1, bf1, cc);   // K = 4..7
      #pragma unroll
      for (int rr = 0; rr < 8; ++rr) {
        int m = mt*16 + hlf*8 + rr;          // C/D layout: lanes16-31 hold rows M=8..15
        out[(size_t)(pb*64 + m)*HWSZ + prem] = cc[rr] + adjb[m];
      }
    }
  }
}

extern "C" void kernel_launch(void* const* d_in, const int* in_sizes, int n_in,
                              void* d_out, int out_size, void* d_ws, size_t ws_size,
                              hipStream_t stream) {
  (void)in_sizes; (void)n_in; (void)out_size; (void)ws_size;
  const float* x    = (const float*)d_in[0];
  const float* gcm  = (const float*)d_in[1];
  const float* cHw  = (const float*)d_in[2];
  const float* cHb  = (const float*)d_in[3];
  const float* cSw  = (const float*)d_in[4];
  const float* cSb  = (const float*)d_in[5];
  const float* cRw  = (const float*)d_in[6];
  const float* cRb  = (const float*)d_in[7];
  // d_in[8], d_in[9] = convWw_* : dead in the reference (Ww features are zeroed)
  const float* adjw = (const float*)d_in[10];
  const float* adjb = (const float*)d_in[11];
  const float* c0w  = (const float*)d_in[12];
  const float* c0b  = (const float*)d_in[13];
  const float* c1w  = (const float*)d_in[14];
  const float* c1b  = (const float*)d_in[15];
  const float* c2w  = (const float*)d_in[16];
  const float* c2b  = (const float*)d_in[17];
  const float* p0w  = (const float*)d_in[18];
  const float* p0b  = (const float*)d_in[19];
  const float* p1w  = (const float*)d_in[20];
  const float* p1b  = (const float*)d_in[21];
  const float* p2w  = (const float*)d_in[22];
  const float* p2b  = (const float*)d_in[23];

  float* ws   = (float*)d_ws;
  float* XG   = ws + XG_OFF;
  float* BUFA = ws + BUFA_OFF;   // s0, then blur-tmp, then wp0 output
  float* BUFB = ws + BUFB_OFF;   // s1, then wp1 output
  float* EB   = ws + EB_OFF;     // blurred E / El / Ell
  float* PART = ws + PART_OFF;
  float* TAPS = ws + TAPS_OFF;
  float* MM   = ws + MM_OFF;
  float* BV   = ws + BIAS_OFF;
  float* outp = (float*)d_out;

  const int PB = NPIX / 256;     // 4608 blocks (also 16x16 tile count)

  k_xg<<<PB, 256, 0, stream>>>(x, gcm, XG);
  k_convw<3, 1><<<PB, 256, 0, stream>>>(x,    c0w, c0b, BUFA);   // silu (WMMA)
  k_convw<16,1><<<PB, 256, 0, stream>>>(BUFA, c1w, c1b, BUFB);   // silu (WMMA + TDM)
  k_c2_reduce<<<NB*576, 256, 0, stream>>>(BUFB, c2w, c2b, PART);
  k_taps<<<1, 256, 0, stream>>>(PART, TAPS);
  k_blur_h<<<NB*3*HH, WW, 0, stream>>>(XG, TAPS, BUFA);          // BUFA reused as tmp
  k_blur_v<<<(NB*3*HWSZ)/256, 256, 0, stream>>>(BUFA, TAPS, EB);
  k_convw<3, 2><<<PB, 256, 0, stream>>>(XG,   p0w, p0b, BUFA);   // relu (WMMA, overwrites tmp)
  k_convw<16,2><<<PB, 256, 0, stream>>>(BUFA, p1w, p1b, BUFB);   // relu (WMMA + TDM)
  k_build_M<<<1, 64, 0, stream>>>(adjw, adjb, cHw, cHb, cSw, cSb, cRw, cRb, MM, BV);
  k_final<<<PB, 256, 0, stream>>>(x, BUFB, EB, MM, BV, p2w, p2b, outp);
}